// Head_9766755631481
// MI455X (gfx1250) — compile-verified
//
#include <hip/hip_runtime.h>
#include <cmath>
#include <stdint.h>

#define BATCH 1024
#define NMEM  4096
#define MDIM  64
#define CDIM  512
#define NOUT  70      // Mdim + 6
#define PPAD  80      // padded params row
#define EPSV  1e-16f

#define TB     128    // threads per block in main kernel
#define CHUNK  128    // mem rows staged per buffer
#define STRIDE 68     // LDS row stride (floats): 64 data + 4 pad DWORDs
#define NCHUNK (NMEM / CHUNK)               // 32
#define LPT    (CHUNK * MDIM / 4 / TB)      // 16 float4 copies per thread

typedef __attribute__((ext_vector_type(2))) float        v2f;
typedef __attribute__((ext_vector_type(8))) float        v8f;
typedef __attribute__((ext_vector_type(4))) int          v4i;
typedef __attribute__((ext_vector_type(8))) int          v8i;
typedef __attribute__((ext_vector_type(4))) unsigned int v4u;

#define AS1 __attribute__((address_space(1)))
#define AS3 __attribute__((address_space(3)))

#if defined(__has_builtin)
#if __has_builtin(__builtin_amdgcn_tensor_load_to_lds) && __has_builtin(__builtin_amdgcn_s_wait_tensorcnt)
#define HAVE_TDM 1
#endif
#if __has_builtin(__builtin_amdgcn_global_load_async_to_lds_b128)
#define HAVE_ASYNC_LDS 1
#endif
#endif

// ---------------------------------------------------------------------------
// TDM: one instruction stages a whole 128x64 f32 tile into LDS with 4-DWORD
// padding every 64 DWORDs (row stride 68 floats) done by the DMA engine.
// D# per CDNA5 ISA 8.3/8.4.
// ---------------------------------------------------------------------------
#if defined(HAVE_TDM)
__device__ __forceinline__ void tdm_load_chunk(const float* gsrc, float* ldsDst)
{
    uint64_t ga = (uint64_t)(uintptr_t)gsrc;
    uint32_t la = (uint32_t)(uintptr_t)ldsDst;   // low 32 bits = LDS byte offset

    v4u g0;
    g0.x = 1u;                                    // count=1 valid, no gather
    g0.y = la;                                    // lds_addr
    g0.z = (uint32_t)ga;                          // global_addr[31:0]
    g0.w = (uint32_t)((ga >> 32) & 0x01FFFFFFu)   // global_addr[56:32]
         | (2u << 30);                            // type = 2 ("image")

    v8i g1;
    g1[0] = (2 << 16)                             // data_size = 4 bytes
          | (1 << 20)                             // pad_enable
          | (5 << 22)                             // pad_interval: 64 DWORDs
          | (3 << 25);                            // pad_amount: 4 DWORDs
    g1[1] = (MDIM & 0xFFFF) << 16;                // tensor_dim0 = 64 (bits 63:48)
    g1[2] = (int)((unsigned)NMEM << 16);          // tensor_dim1 lo16 (bits 95:80)
    g1[3] = (MDIM & 0xFFFF) << 16;                // tile_dim0 = 64 (bits 127:112)
    g1[4] = CHUNK;                                // tile_dim1 = 128, tile_dim2 = 0
    g1[5] = MDIM;                                 // tensor_dim0_stride = 64
    g1[6] = 0;
    g1[7] = 0;

    v4i g2 = {0, 0, 0, 0};                        // unused (<=2D tensor)
    v4i g3 = {0, 0, 0, 0};
    v8i g4 = {0, 0, 0, 0, 0, 0, 0, 0};            // extra group (6-arg form)

    __builtin_amdgcn_tensor_load_to_lds(g0, g1, g2, g3, g4, 0);
}
__device__ __forceinline__ void wait_tdm()
{
    __builtin_amdgcn_s_wait_tensorcnt(0);
}
#endif

// per-lane async (or fallback sync) 16-byte global -> LDS copy
__device__ __forceinline__ void copy16_to_lds(const float* src, float* dstLds)
{
#if defined(HAVE_ASYNC_LDS)
    AS3 v4i* d = (AS3 v4i*)(uint32_t)(uintptr_t)dstLds;
    AS1 v4i* s = (AS1 v4i*)(uintptr_t)src;
    __builtin_amdgcn_global_load_async_to_lds_b128(s, d, 0, 0);
#else
    *(float4*)dstLds = *(const float4*)src;
#endif
}

__device__ __forceinline__ void wait_async_lds()
{
#if defined(HAVE_ASYNC_LDS)
#if __has_builtin(__builtin_amdgcn_s_wait_asynccnt)
    __builtin_amdgcn_s_wait_asynccnt(0);
#else
    asm volatile("s_wait_asynccnt 0x0" ::: "memory");
#endif
#endif
}

// ---------------------------------------------------------------------------
// Kernel 1: params = controller_out @ W^T + b  via V_WMMA_F32_16X16X4_F32
// one wave per 16x16 tile; grid = (B/16, ceil(70/16)=5). Branch-free B masking.
// ---------------------------------------------------------------------------
__global__ __launch_bounds__(32) void ntm_params_wmma(
    const float* __restrict__ A,     // (B, C)
    const float* __restrict__ W,     // (70, C)
    const float* __restrict__ bias,  // (70)
    float* __restrict__ params)      // (B, 80)
{
    const int lane = threadIdx.x;
    const int tm   = blockIdx.x;         // 0..63
    const int tn   = blockIdx.y;         // 0..4
    const int half = lane >> 4;          // 0: lanes 0-15, 1: lanes 16-31
    const int l16  = lane & 15;
    const int rowA = tm * 16 + l16;      // M index for A fragment
    const int colN = tn * 16 + l16;      // N index for B fragment
    const bool  colOK = (colN < NOUT);
    const float msk   = colOK ? 1.0f : 0.0f;

    const float* aPtr = A + (size_t)rowA * CDIM + half * 2;
    const float* wPtr = W + (size_t)(colOK ? colN : (NOUT - 1)) * CDIM + half * 2;

    v8f c = {};
    for (int kk = 0; kk < CDIM; kk += 4) {
        v2f a  = *(const v2f*)(aPtr + kk);
        v2f bf = *(const v2f*)(wPtr + kk);
        bf.x *= msk;
        bf.y *= msk;
        c = __builtin_amdgcn_wmma_f32_16x16x4_f32(
                false, a, false, bf, (short)0, c, false, false);
    }

    const int rowBase = tm * 16 + half * 8;
    const float bv = colOK ? bias[colN] : 0.0f;
#pragma unroll
    for (int i = 0; i < 8; ++i) {
        params[(size_t)(rowBase + i) * PPAD + colN] = c[i] + bv;
    }
}

// ---------------------------------------------------------------------------
// Kernel 2: per-batch activations -> scal[b] = {beta, g, s0, s1, s2, gamma, |k|}
// ---------------------------------------------------------------------------
__global__ void ntm_activations(const float* __restrict__ params,
                                float* __restrict__ scal)
{
    int b = blockIdx.x * blockDim.x + threadIdx.x;
    if (b >= BATCH) return;
    const float* p = params + (size_t)b * PPAD;
    float x0 = p[MDIM + 0], x1 = p[MDIM + 1], x2 = p[MDIM + 2];
    float x3 = p[MDIM + 3], x4 = p[MDIM + 4], x5 = p[MDIM + 5];

    float beta = (x0 > 20.0f) ? x0 : log1pf(expf(x0));     // softplus
    float g    = 1.0f / (1.0f + expf(-x1));                // sigmoid
    float mx   = fmaxf(x2, fmaxf(x3, x4));
    float e0 = expf(x2 - mx), e1 = expf(x3 - mx), e2 = expf(x4 - mx);
    float es = e0 + e1 + e2;
    float gamma = fmaxf(x5, 0.0f) + 1.0f;                  // relu + 1

    float nsq = 0.0f;
#pragma unroll 8
    for (int m = 0; m < MDIM; ++m) { float v = p[m]; nsq = fmaf(v, v, nsq); }

    float* s = scal + (size_t)b * 8;
    s[0] = beta; s[1] = g; s[2] = e0 / es; s[3] = e1 / es; s[4] = e2 / es;
    s[5] = gamma; s[6] = sqrtf(nsq);
}

// ---------------------------------------------------------------------------
// Kernel 3: fused streaming kernel, one block per batch row.
// TDM (or async-to-LDS) double-buffered staging of mem; cosine sim ->
// softmax -> interpolation -> circular conv -> sharpen -> normalize, in-LDS.
// ---------------------------------------------------------------------------
__global__ __launch_bounds__(TB) void ntm_address(
    const float* __restrict__ mem,     // (B, N, M)
    const float* __restrict__ prev_w,  // (B, N)
    const float* __restrict__ params,  // (B, 80) : k in [0,64)
    const float* __restrict__ scal,    // (B, 8)
    float* __restrict__ out)           // (B, N)
{
    __shared__ __align__(16) float STAGE[2][CHUNK * STRIDE]; // 2 x 34,816 B
    __shared__ float ZE[NMEM];                               // 16,384 B
    __shared__ float KS[MDIM];
    __shared__ float SC[8];
    __shared__ float red[8];

    const int t    = threadIdx.x;
    const int b    = blockIdx.x;
    const int lane = t & 31;
    const int wid  = t >> 5;

    if (t < MDIM) KS[t] = params[(size_t)b * PPAD + t];
    if (t < 7)    SC[t] = scal[(size_t)b * 8 + t];
    __syncthreads();
    const float beta = SC[0], g = SC[1], s0 = SC[2], s1 = SC[3], s2 = SC[4];
    const float gamma = SC[5], knorm = SC[6];

    const float* memB = mem + (size_t)b * NMEM * MDIM;
    float zmaxLocal = -INFINITY;

    // ---- prologue: stage chunk 0 into buffer 0 -----------------------------
#if defined(HAVE_TDM)
    if (wid == 0) tdm_load_chunk(memB, &STAGE[0][0]);
#else
#pragma unroll
    for (int i = 0; i < LPT; ++i) {
        int v = i * TB + t;
        int r = v >> 4;
        int c = (v & 15) * 4;
        copy16_to_lds(memB + (size_t)v * 4, &STAGE[0][r * STRIDE + c]);
    }
#endif

    // ---- Phase A: pipelined stream of mem (dot + norm per row) -------------
    for (int chunk = 0; chunk < NCHUNK; ++chunk) {
        const int cur = chunk & 1;
#if defined(HAVE_TDM)
        if (wid == 0) wait_tdm();     // issuing wave's TDM complete
#else
        wait_async_lds();             // this wave's staged loads complete
#endif
        __syncthreads();              // all waves see staged data; prev compute done

        if (chunk + 1 < NCHUNK) {
            const float* srcn = memB + (size_t)(chunk + 1) * CHUNK * MDIM;
            if (chunk + 2 < NCHUNK)
                __builtin_prefetch(memB + (size_t)(chunk + 2) * CHUNK * MDIM + t * 16, 0, 0);
#if defined(HAVE_TDM)
            if (wid == 0) tdm_load_chunk(srcn, &STAGE[cur ^ 1][0]);
#else
#pragma unroll
            for (int i = 0; i < LPT; ++i) {
                int v = i * TB + t;
                int r = v >> 4;
                int c = (v & 15) * 4;
                copy16_to_lds(srcn + (size_t)v * 4, &STAGE[cur ^ 1][r * STRIDE + c]);
            }
#endif
        }

        float dot = 0.0f, nsq = 0.0f;
        const float* rowp = &STAGE[cur][t * STRIDE];
#pragma unroll
        for (int m = 0; m < MDIM; ++m) {
            float v = rowp[m];
            dot = fmaf(v, KS[m], dot);
            nsq = fmaf(v, v, nsq);
        }
        float sim = dot / (sqrtf(nsq) * knorm + EPSV);
        float z = beta * sim;
        ZE[chunk * CHUNK + t] = z;
        zmaxLocal = fmaxf(zmaxLocal, z);
        // no trailing barrier: next iteration's wait+barrier guards the buffer
    }

    // ---- block max(z) ------------------------------------------------------
    float m = zmaxLocal;
    for (int o = 16; o > 0; o >>= 1) m = fmaxf(m, __shfl_xor(m, o, 32));
    if (lane == 0) red[wid] = m;
    __syncthreads();
    if (t == 0) {
        float mm = red[0];
        for (int i = 1; i < TB / 32; ++i) mm = fmaxf(mm, red[i]);
        red[0] = mm;
    }
    __syncthreads();
    const float zmax = red[0];
    __syncthreads();

    // ---- softmax: e = exp(z - max), sum ------------------------------------
    float sumLocal = 0.0f;
    for (int j = 0; j < NMEM / TB; ++j) {
        int n = j * TB + t;
        float e = expf(ZE[n] - zmax);
        ZE[n] = e;
        sumLocal += e;
    }
    float s = sumLocal;
    for (int o = 16; o > 0; o >>= 1) s += __shfl_xor(s, o, 32);
    if (lane == 0) red[wid] = s;
    __syncthreads();
    if (t == 0) {
        float ss = 0.0f;
        for (int i = 0; i < TB / 32; ++i) ss += red[i];
        red[0] = ss;
    }
    __syncthreads();
    const float invS = 1.0f / red[0];
    __syncthreads();

    // ---- gate with prev_w: w_g = g*w_c + (1-g)*prev_w ----------------------
    const float* pw = prev_w + (size_t)b * NMEM;
    for (int j = 0; j < NMEM / TB; ++j) {
        int n = j * TB + t;
        float wc = ZE[n] * invS;
        ZE[n] = g * wc + (1.0f - g) * pw[n];
    }
    __syncthreads();

    // ---- circular 3-tap shift + sharpen ------------------------------------
    float wsr[NMEM / TB];
    float sum2Local = 0.0f;
#pragma unroll
    for (int j = 0; j < NMEM / TB; ++j) {
        int n   = j * TB + t;
        int nm1 = (n + NMEM - 1) & (NMEM - 1);
        int np1 = (n + 1) & (NMEM - 1);
        float wt  = s0 * ZE[nm1] + s1 * ZE[n] + s2 * ZE[np1];
        float wsh = powf(wt, gamma);
        wsr[j] = wsh;
        sum2Local += wsh;
    }
    float s2r = sum2Local;
    for (int o = 16; o > 0; o >>= 1) s2r += __shfl_xor(s2r, o, 32);
    if (lane == 0) red[wid] = s2r;
    __syncthreads();
    if (t == 0) {
        float ss = 0.0f;
        for (int i = 0; i < TB / 32; ++i) ss += red[i];
        red[0] = ss;
    }
    __syncthreads();
    const float inv2 = 1.0f / (red[0] + EPSV);

    float* ob = out + (size_t)b * NMEM;
#pragma unroll
    for (int j = 0; j < NMEM / TB; ++j) {
        ob[j * TB + t] = wsr[j] * inv2;
    }
}

// ---------------------------------------------------------------------------
extern "C" void kernel_launch(void* const* d_in, const int* in_sizes, int n_in,
                              void* d_out, int out_size, void* d_ws, size_t ws_size,
                              hipStream_t stream)
{
    const float* controller = (const float*)d_in[0];   // (1024, 512)
    const float* W          = (const float*)d_in[1];   // (70, 512)
    const float* bias       = (const float*)d_in[2];   // (70)
    const float* mem        = (const float*)d_in[3];   // (1024, 4096, 64)
    const float* prev_w     = (const float*)d_in[4];   // (1024, 4096)
    float* out = (float*)d_out;                        // (1024, 4096)

    float* params = (float*)d_ws;                      // 1024*80 floats
    float* scal   = params + (size_t)BATCH * PPAD;     // 1024*8 floats

    ntm_params_wmma<<<dim3(BATCH / 16, 5), 32, 0, stream>>>(controller, W, bias, params);
    ntm_activations<<<BATCH / 128, 128, 0, stream>>>(params, scal);
    ntm_address<<<BATCH, TB, 0, stream>>>(mem, prev_w, params, scal, out);
}